// QuantumLayer_65498251264457
// MI455X (gfx1250) — compile-verified
//
#include <hip/hip_runtime.h>
#include <math.h>

typedef __attribute__((ext_vector_type(2))) float v2f;
typedef __attribute__((ext_vector_type(8))) float v8f;

#define NQ 13
#define DIM 8192          // 2^13
#define NLAYERS 3
#define TPB 256           // 8 wave32s

// ---------------------------------------------------------------------------
// Kernel 1: build the 3 layers x 3 groups of 16x16 Kronecker RY matrices
// into d_ws, plus cos/sin of qubit-12 angle per layer.
// ws layout: float g[3][3][256] (offset l*768+g*256 + r*16+c), then
//            c12[3] at 2304, s12[3] at 2307.
// ---------------------------------------------------------------------------
__global__ void __launch_bounds__(TPB) build_gates(const float* __restrict__ w,
                                                   float* __restrict__ ws) {
  int t = threadIdx.x;          // 0..255 -> (r,c) of the 16x16 matrices
  int r = t >> 4;
  int c = t & 15;
  for (int l = 0; l < NLAYERS; ++l) {
    for (int g = 0; g < 3; ++g) {
      float v = 1.0f;
#pragma unroll
      for (int j = 0; j < 4; ++j) {
        float ang = w[l * NQ + g * 4 + j] * 0.5f;
        float ch = cosf(ang), sh = sinf(ang);
        int rb = (r >> (3 - j)) & 1;
        int cb = (c >> (3 - j)) & 1;
        // RY = [[c,-s],[s,c]]
        float e = rb ? (cb ? ch : sh) : (cb ? -sh : ch);
        v *= e;
      }
      ws[l * 768 + g * 256 + r * 16 + c] = v;
    }
  }
  if (t < NLAYERS) {
    float ang = w[t * NQ + 12] * 0.5f;
    ws[2304 + t] = cosf(ang);
    ws[2307 + t] = sinf(ang);
  }
}

// ---------------------------------------------------------------------------
// Kernel 2: one block per batch item; state lives in LDS.
//  - build product state from encoding angles (no gates needed)
//  - per layer: 3 in-place WMMA gate applications (qubits 0-3,4-7,8-11),
//    then RY on qubit 12 fused with the CNOT-chain permutation
//    new[j] = ry12(old)[j ^ (j>>1)]  (register-staged, single LDS buffer)
//  - Z expectations via wave reduction + LDS atomics
// ---------------------------------------------------------------------------
__global__ void __launch_bounds__(TPB) qsim(const float* __restrict__ x,
                                            const float* __restrict__ ws,
                                            float* __restrict__ out) {
  __shared__ float st[DIM];        // 32 KB state
  __shared__ float gm[2304];       // 9 KB gate matrices
  __shared__ float enc[2 * NQ];    // cos/sin encoding per qubit
  __shared__ float eacc[NQ];

  const int t = threadIdx.x;
  const int b = blockIdx.x;
  const int lane = t & 31;
  const int wv = t >> 5;

  // stage gates into LDS
  for (int i = t; i < 2304; i += TPB) gm[i] = ws[i];
  if (t < NQ) {
    float h = x[b * NQ + t] * 1.5707963267948966f;
    enc[t] = cosf(h);
    enc[NQ + t] = sinf(h);
    eacc[t] = 0.0f;
  }
  __syncthreads();

  // ---- initial product state: i = t*32 + k ; qubit q lives at bit (12-q)
  float pre = 1.0f;
#pragma unroll
  for (int q = 0; q < 8; ++q) {
    int bit = (t >> (7 - q)) & 1;       // bits 12..5 of i come from t
    pre *= bit ? enc[NQ + q] : enc[q];
  }
#pragma unroll 1
  for (int k = 0; k < 32; ++k) {
    float v = pre;
#pragma unroll
    for (int q = 8; q < 13; ++q) {
      int bit = (k >> (12 - q)) & 1;    // bits 4..0 of i come from k
      v *= bit ? enc[NQ + q] : enc[q];
    }
    st[t * 32 + k] = v;
  }
  __syncthreads();

  const int m = lane & 15;              // matrix row (A) / column (B,D)
  const int koff = (lane >> 4) << 1;    // 0 or 2 : K split across lane halves
  const int rbase = (lane >> 4) << 3;   // 0 or 8 : D row split

#pragma unroll 1
  for (int layer = 0; layer < NLAYERS; ++layer) {
    // ---- three grouped-RY 16x16 matrix applications (in-place, WMMA fp32)
#pragma unroll 1
    for (int g = 0; g < 3; ++g) {
      const float* G = &gm[layer * 768 + g * 256];
      // A operand: lane m holds G[m][kbase+koff], G[m][kbase+koff+1]
      float a[8];
#pragma unroll
      for (int kk = 0; kk < 4; ++kk) {
        a[2 * kk]     = G[m * 16 + 4 * kk + koff];
        a[2 * kk + 1] = G[m * 16 + 4 * kk + koff + 1];
      }
      const int s = 9 - 4 * g;               // fiber bit position
      const int lowmask = (1 << s) - 1;
      // each tile iteration is self-contained: keep it rolled so only one
      // tile's addresses are live (stay under 256 VGPRs, no vgpr-msb churn)
#pragma unroll 1
      for (int tt = 0; tt < 4; ++tt) {       // 4 tiles per wave, 32 total
        int col = (wv * 4 + tt) * 16 + m;    // 0..511
        int base = ((col >> s) << (s + 4)) | (col & lowmask);
        v8f acc = {0.f, 0.f, 0.f, 0.f, 0.f, 0.f, 0.f, 0.f};
#pragma unroll
        for (int kk = 0; kk < 4; ++kk) {
          int r0 = 4 * kk + koff;
          v2f bv;
          bv.x = st[base | (r0 << s)];
          bv.y = st[base | ((r0 + 1) << s)];
          v2f av;
          av.x = a[2 * kk];
          av.y = a[2 * kk + 1];
          acc = __builtin_amdgcn_wmma_f32_16x16x4_f32(
              /*neg_a=*/false, av, /*neg_b=*/false, bv,
              /*c_mod=*/(short)0, acc, /*reuse_a=*/false, /*reuse_b=*/false);
        }
#pragma unroll
        for (int j = 0; j < 8; ++j)
          st[base | ((rbase + j) << s)] = acc[j];
      }
      __syncthreads();
    }

    // ---- qubit-12 RY fused with CNOT-chain permutation new[j]=ry[j^(j>>1)]
    float cc = ws[2304 + layer];
    float ssn = ws[2307 + layer];
    float tmp[32];                    // must be fully register-resident:
#pragma unroll                        // all reads happen before any write
    for (int k = 0; k < 32; ++k) {
      int j = t * 32 + k;
      int src = j ^ (j >> 1);
      float s0 = st[src & ~1];
      float s1 = st[src | 1];
      tmp[k] = (src & 1) ? fmaf(ssn, s0, cc * s1)
                         : fmaf(cc, s0, -ssn * s1);
    }
    __syncthreads();
#pragma unroll
    for (int k = 0; k < 32; ++k) st[t * 32 + k] = tmp[k];
    __syncthreads();
  }

  // ---- Z expectations: exp_q = sum_i (+-) state[i]^2
  float acc13[NQ];
#pragma unroll
  for (int q = 0; q < NQ; ++q) acc13[q] = 0.0f;
#pragma unroll 1
  for (int k = 0; k < 32; ++k) {
    int i = t * 32 + k;
    float v = st[i];
    float p = v * v;
#pragma unroll
    for (int q = 0; q < NQ; ++q)
      acc13[q] += ((i >> (12 - q)) & 1) ? -p : p;
  }
#pragma unroll
  for (int q = 0; q < NQ; ++q) {
#pragma unroll
    for (int off = 16; off > 0; off >>= 1)
      acc13[q] += __shfl_xor(acc13[q], off, 32);
  }
  if (lane == 0) {
#pragma unroll
    for (int q = 0; q < NQ; ++q) atomicAdd(&eacc[q], acc13[q]);
  }
  __syncthreads();
  if (t < NQ) out[b * NQ + t] = eacc[t];
}

// ---------------------------------------------------------------------------
extern "C" void kernel_launch(void* const* d_in, const int* in_sizes, int n_in,
                              void* d_out, int out_size, void* d_ws, size_t ws_size,
                              hipStream_t stream) {
  const float* x = (const float*)d_in[0];        // (B, 13) f32
  const float* w = (const float*)d_in[1];        // (39,)   f32
  float* out = (float*)d_out;                    // (B, 13) f32
  float* ws = (float*)d_ws;                      // >= 2310 floats

  int B = in_sizes[0] / NQ;                      // 1024

  build_gates<<<1, TPB, 0, stream>>>(w, ws);
  qsim<<<B, TPB, 0, stream>>>(x, ws, out);
}